// DecoderBlock_34239479284315
// MI455X (gfx1250) — compile-verified
//
#include <hip/hip_runtime.h>
#include <hip/hip_bf16.h>
#include <math.h>

#define VOCAB 32000
#define EMB   512
#define TT    64
#define SS    64
#define UNITS 1024
#define BB    32

#define GATES (4 * UNITS)     // 4096
#define XDIM  (EMB + UNITS)   // 1536
#define KC    256             // K chunk staged in LDS
#define KCP   257             // padded row stride (bank-conflict free)

typedef float v2f __attribute__((ext_vector_type(2)));
typedef float v8f __attribute__((ext_vector_type(8)));

#define WMMA_F32(A, Bf, C) \
  __builtin_amdgcn_wmma_f32_16x16x4_f32(false, (A), false, (Bf), (short)0, (C), false, false)

// ---------------------------------------------------------------------------
// Cooperative load of a 32 x 256 K-chunk of the A matrix into padded LDS.
// A is row-major [32 x K]. 256 threads, coalesced along K.
// ---------------------------------------------------------------------------
__device__ __forceinline__ void load_a_chunk(float* __restrict__ sh,
                                             const float* __restrict__ A,
                                             int K, int kbase, int tid) {
#pragma unroll
  for (int i = tid; i < 32 * KC; i += 256) {
    int r = i >> 8;          // row 0..31
    int cc = i & (KC - 1);   // col 0..255
    sh[r * KCP + cc] = A[r * K + kbase + cc];
  }
}

// ---------------------------------------------------------------------------
// One K-chunk with 2x2 register blocking: per k-step, 2 A frags (rows 0-15 and
// 16-31) x 2 B frags (cols n0, n0+16) -> 4 x v_wmma_f32_16x16x4_f32.
// Fragment layout (ISA 7.12.2): lanes 0-15 hold K=k..k+1, lanes 16-31 K=k+2..k+3.
// ---------------------------------------------------------------------------
__device__ __forceinline__ void wmma_chunk_2x2(const float* __restrict__ sh,
                                               const float* __restrict__ Bcol,
                                               int kbase, int N,
                                               int lh, int hi,
                                               v8f& c00, v8f& c01,
                                               v8f& c10, v8f& c11) {
#pragma unroll 4
  for (int k = 0; k < KC; k += 4) {
    int kk = k + 2 * hi;
    v2f a0, a1, b0, b1;
    const float* ap0 = &sh[lh * KCP + kk];
    const float* ap1 = &sh[(16 + lh) * KCP + kk];
    a0.x = ap0[0]; a0.y = ap0[1];
    a1.x = ap1[0]; a1.y = ap1[1];
    const float* bp = Bcol + (size_t)(kbase + kk) * (size_t)N;
    b0.x = bp[0];          b0.y = bp[(size_t)N];
    b1.x = bp[16];         b1.y = bp[(size_t)N + 16];
    c00 = WMMA_F32(a0, b0, c00);
    c01 = WMMA_F32(a0, b1, c01);
    c10 = WMMA_F32(a1, b0, c10);
    c11 = WMMA_F32(a1, b1, c11);
  }
}

// D layout (ISA 7.12.2): VGPR v, lanes 0-15 -> row m0+v, lanes 16-31 -> row m0+v+8.
__device__ __forceinline__ void store_tile(float* __restrict__ base,
                                           size_t rowStride, int m0, int col,
                                           int hi, const v8f& acc, float bias) {
#pragma unroll
  for (int v = 0; v < 8; ++v) {
    int row = m0 + v + hi * 8;
    base[(size_t)row * rowStride + col] = acc[v] + bias;
  }
}

// ---------------------------------------------------------------------------
// Zero-init recurrent state (h, c) — required every call (graph replay).
// ---------------------------------------------------------------------------
__global__ void init_state_kernel(float* __restrict__ p, int n) {
  for (int i = blockIdx.x * blockDim.x + threadIdx.x; i < n;
       i += gridDim.x * blockDim.x)
    p[i] = 0.0f;
}

// ---------------------------------------------------------------------------
// Attention + embedding gather. One block per batch element (32 blocks, 256thr).
//   scores = h . enc[b,s,:]  -> softmax -> ctx = attn . enc
//   x[b] = [ embedding[tok[b,t]] ; ctx ]
// ---------------------------------------------------------------------------
__global__ void attn_embed_kernel(const int* __restrict__ tok,
                                  const float* __restrict__ enc,
                                  const float* __restrict__ emb_table,
                                  const float* __restrict__ h,
                                  float* __restrict__ x, int t) {
  const int b = blockIdx.x;
  const int tid = threadIdx.x;
  const int lane = tid & 31;
  const int w = tid >> 5;

  __shared__ float sc[SS];

  const float* hb = h + b * UNITS;
  const float* encb = enc + (size_t)b * SS * UNITS;

  // scores[s] = sum_u h[u] * enc[b][s][u] ; wave w handles s = w, w+8, ...
  for (int s = w; s < SS; s += 8) {
    const float* e = encb + (size_t)s * UNITS;
    float p = 0.f;
    for (int u = lane; u < UNITS; u += 32) p = fmaf(hb[u], e[u], p);
#pragma unroll
    for (int off = 16; off > 0; off >>= 1) p += __shfl_xor(p, off, 32);
    if (lane == 0) sc[s] = p;
  }
  __syncthreads();

  // softmax over S=64 (tiny; serialized, 32 blocks total so negligible)
  if (tid == 0) {
    float m = sc[0];
    for (int s = 1; s < SS; ++s) m = fmaxf(m, sc[s]);
    float sum = 0.f;
    for (int s = 0; s < SS; ++s) { float e = expf(sc[s] - m); sc[s] = e; sum += e; }
    float inv = 1.f / sum;
    for (int s = 0; s < SS; ++s) sc[s] *= inv;
  }
  __syncthreads();

  float* xb = x + b * XDIM;

  // embedding gather: x[b][0:512]
  const int token = tok[b * TT + t];
  const float* er = emb_table + (size_t)token * EMB;
  for (int i = tid; i < EMB; i += 256) xb[i] = er[i];

  // context: x[b][512 : 512+1024], coalesced over u
  for (int u = tid; u < UNITS; u += 256) {
    float acc = 0.f;
    const float* e = encb + u;
#pragma unroll 8
    for (int s = 0; s < SS; ++s) acc = fmaf(sc[s], e[(size_t)s * UNITS], acc);
    xb[EMB + u] = acc;
  }
}

// ---------------------------------------------------------------------------
// Gate GEMM: z[32 x 4096] = x[32x1536] @ Wx + h[32x1024] @ Wh + b_lstm
// grid = 16 blocks (each block: 256 output columns), 256 threads = 8 waves,
// each wave computes a 32x32 output tile (2x2 WMMA register blocking).
// ---------------------------------------------------------------------------
__global__ void gates_gemm_kernel(const float* __restrict__ x,
                                  const float* __restrict__ h,
                                  const float* __restrict__ Wx,
                                  const float* __restrict__ Wh,
                                  const float* __restrict__ bl,
                                  float* __restrict__ z) {
  __shared__ float sh[32 * KCP];
  const int tid = threadIdx.x;
  const int w = tid >> 5;
  const int lane = tid & 31;
  const int lh = lane & 15;
  const int hi = lane >> 4;
  const int n0 = blockIdx.x * 256 + w * 32;

  v8f c00 = {}, c01 = {}, c10 = {}, c11 = {};

  // phase 1: x @ Wx, K = 1536
  const float* BcolX = Wx + n0 + lh;
  for (int kc = 0; kc < XDIM / KC; ++kc) {
    load_a_chunk(sh, x, XDIM, kc * KC, tid);
    __syncthreads();
    wmma_chunk_2x2(sh, BcolX, kc * KC, GATES, lh, hi, c00, c01, c10, c11);
    __syncthreads();
  }
  // phase 2: h @ Wh, K = 1024
  const float* BcolH = Wh + n0 + lh;
  for (int kc = 0; kc < UNITS / KC; ++kc) {
    load_a_chunk(sh, h, UNITS, kc * KC, tid);
    __syncthreads();
    wmma_chunk_2x2(sh, BcolH, kc * KC, GATES, lh, hi, c00, c01, c10, c11);
    __syncthreads();
  }

  const int col0 = n0 + lh;
  const int col1 = n0 + 16 + lh;
  store_tile(z, GATES, 0,  col0, hi, c00, bl[col0]);
  store_tile(z, GATES, 0,  col1, hi, c01, bl[col1]);
  store_tile(z, GATES, 16, col0, hi, c10, bl[col0]);
  store_tile(z, GATES, 16, col1, hi, c11, bl[col1]);
}

// ---------------------------------------------------------------------------
// LSTM pointwise update: gates in z -> new (h, c), in place in workspace.
// ---------------------------------------------------------------------------
__global__ void lstm_update_kernel(const float* __restrict__ z,
                                   float* __restrict__ h,
                                   float* __restrict__ c) {
  const int idx = blockIdx.x * 256 + threadIdx.x;  // 0 .. 32767
  const int b = idx >> 10;
  const int u = idx & (UNITS - 1);
  const float* zb = z + b * GATES;
  float gi = zb[u];
  float gf = zb[UNITS + u];
  float gg = zb[2 * UNITS + u];
  float go = zb[3 * UNITS + u];
  float si = 1.f / (1.f + expf(-gi));
  float sf = 1.f / (1.f + expf(-gf));
  float so = 1.f / (1.f + expf(-go));
  float cn = sf * c[idx] + si * tanhf(gg);
  float hn = so * tanhf(cn);
  c[idx] = cn;
  h[idx] = hn;
}

// ---------------------------------------------------------------------------
// Logits GEMM: out[:, t, :] = h[32x1024] @ Wo[1024x32000] + bo
// grid = 125 blocks x 256 columns; Wo (131 MB fp32) is L2-resident after
// step 0 (192 MB global L2), so steps 1..63 run at L2 bandwidth.
// ---------------------------------------------------------------------------
__global__ void logits_gemm_kernel(const float* __restrict__ h,
                                   const float* __restrict__ Wo,
                                   const float* __restrict__ bo,
                                   float* __restrict__ out, int t) {
  __shared__ float sh[32 * KCP];
  const int tid = threadIdx.x;
  const int w = tid >> 5;
  const int lane = tid & 31;
  const int lh = lane & 15;
  const int hi = lane >> 4;
  const int n0 = blockIdx.x * 256 + w * 32;

  v8f c00 = {}, c01 = {}, c10 = {}, c11 = {};

  const float* Bcol = Wo + n0 + lh;
  for (int kc = 0; kc < UNITS / KC; ++kc) {
    load_a_chunk(sh, h, UNITS, kc * KC, tid);
    __syncthreads();
    wmma_chunk_2x2(sh, Bcol, kc * KC, VOCAB, lh, hi, c00, c01, c10, c11);
    __syncthreads();
  }

  float* obase = out + (size_t)t * VOCAB;          // row stride TT*VOCAB (batch)
  const size_t rs = (size_t)TT * VOCAB;
  const int col0 = n0 + lh;
  const int col1 = n0 + 16 + lh;
  store_tile(obase, rs, 0,  col0, hi, c00, bo[col0]);
  store_tile(obase, rs, 0,  col1, hi, c01, bo[col1]);
  store_tile(obase, rs, 16, col0, hi, c10, bo[col0]);
  store_tile(obase, rs, 16, col1, hi, c11, bo[col1]);
}

// ---------------------------------------------------------------------------
// Host-side orchestration: 64 sequential timesteps, 4 kernels each.
// ---------------------------------------------------------------------------
extern "C" void kernel_launch(void* const* d_in, const int* in_sizes, int n_in,
                              void* d_out, int out_size, void* d_ws, size_t ws_size,
                              hipStream_t stream) {
  const int*   tok  = (const int*)d_in[0];
  const float* enc  = (const float*)d_in[1];
  const float* embt = (const float*)d_in[2];
  const float* Wx   = (const float*)d_in[3];
  const float* Wh   = (const float*)d_in[4];
  const float* bl   = (const float*)d_in[5];
  const float* Wo   = (const float*)d_in[6];
  const float* bo   = (const float*)d_in[7];
  float* out = (float*)d_out;

  float* ws = (float*)d_ws;
  float* h = ws;                      // 32*1024
  float* c = h + BB * UNITS;          // 32*1024
  float* x = c + BB * UNITS;          // 32*1536
  float* z = x + BB * XDIM;           // 32*4096

  // zero recurrent state every call (deterministic across graph replays)
  init_state_kernel<<<64, 256, 0, stream>>>(h, 2 * BB * UNITS);

  for (int t = 0; t < TT; ++t) {
    attn_embed_kernel<<<BB, 256, 0, stream>>>(tok, enc, embt, h, x, t);
    gates_gemm_kernel<<<GATES / 256, 256, 0, stream>>>(x, h, Wx, Wh, bl, z);
    lstm_update_kernel<<<(BB * UNITS) / 256, 256, 0, stream>>>(z, h, c);
    logits_gemm_kernel<<<VOCAB / 256, 256, 0, stream>>>(h, Wo, bo, out, t);
  }
}